// ProgramGNN_34668976013873
// MI455X (gfx1250) — compile-verified
//
#include <hip/hip_runtime.h>
#include <hip/hip_bf16.h>
#include <stdint.h>

// ---------------- problem constants (match reference) ----------------
#define Nn   100000
#define Ee   600000
#define Hh   128
#define BCc  384
#define NTt  6250            // Nn / 16 node tiles (exact)
#define NEG_SLOPE 0.01f

typedef __attribute__((ext_vector_type(2))) float v2f;
typedef __attribute__((ext_vector_type(8))) float v8f;

// Exact-fp32 WMMA: D = A(16x4) * B(4x16) + C  (V_WMMA_F32_16X16X4_F32)
__device__ __forceinline__ v8f wmma4(v2f a, v2f b, v8f c) {
  return __builtin_amdgcn_wmma_f32_16x16x4_f32(
      /*neg_a=*/false, a, /*neg_b=*/false, b,
      /*c_mod=*/(short)0, c, /*reuse_a=*/false, /*reuse_b=*/false);
}

// Async memory -> LDS (GLOBAL_LOAD_ASYNC_TO_LDS_B128, tracked by ASYNCcnt).
// vdst = per-lane LDS byte offset (low 32 bits of the generic pointer),
// vaddr = per-lane 64-bit global address, GV mode (saddr = off).
__device__ __forceinline__ void async_g2l_b128(const float* lds_dst, const float* gsrc) {
  uint32_t l = (uint32_t)(uintptr_t)lds_dst;
  asm volatile("global_load_async_to_lds_b128 %0, %1, off"
               :: "v"(l), "v"(gsrc) : "memory");
}

// acc += ascale*A_tile(16 x K, LDS, ld=lda) @ W[:, n0:n0+16] (W row-major, 128 cols).
// Row scaling folds into the A fragment: lane l owns row M = l&15 only.
// A frag: lane l holds (M=l&15, K=k+(l>>4)*2+{0,1}); B frag: (K=k+(l>>4)*2+{0,1}, N=l&15)
__device__ __forceinline__ v8f gemm_tile(const float* __restrict__ As, int lda,
                                         const float* __restrict__ W, int K,
                                         int n0, v8f acc, int lo, int hi, float ascale) {
#pragma unroll 4
  for (int k = 0; k < K; k += 4) {
    const int ka = k + (hi << 1);
    v2f a; a.x = As[lo * lda + ka] * ascale;  a.y = As[lo * lda + ka + 1] * ascale;
    v2f b; b.x = W[ka * Hh + n0 + lo];        b.y = W[(ka + 1) * Hh + n0 + lo];
    acc = wmma4(a, b, acc);
  }
  return acc;
}

// ---------------- utility kernels ----------------
__global__ void zero_kernel(float* __restrict__ p, long n) {
  long i = (long)blockIdx.x * blockDim.x + threadIdx.x;
  long stride = (long)gridDim.x * blockDim.x;
  for (; i < n; i += stride) p[i] = 0.0f;
}

// deg[dst] (edge in-degree) and ccnt[c] (cluster sizes) -- static across layers
__global__ void counts_kernel(const int* __restrict__ e, const int* __restrict__ cc,
                              float* __restrict__ deg, float* __restrict__ ccnt) {
  int t = blockIdx.x * blockDim.x + threadIdx.x;
  if (t < Ee) {
    atomicAdd(&deg[e[Ee + t]], 1.0f);          // e[1][t] = dst
  } else if (t < Ee + Nn) {
    atomicAdd(&ccnt[cc[t - Ee]], 1.0f);
  }
}

// ---------------- encoder: x = [cf | slf | z] @ W_enc + b_enc ----------------
__global__ void encoder_kernel(const float* __restrict__ cf, const float* __restrict__ slf,
                               const float* __restrict__ z, const float* __restrict__ W_enc,
                               const float* __restrict__ b_enc, float* __restrict__ x) {
  extern __shared__ float lds[];
  const int lane = threadIdx.x & 31, wv = threadIdx.x >> 5;
  const int tile = blockIdx.x * (blockDim.x >> 5) + wv;
  if (tile >= NTt) return;                     // wave-uniform: EXEC stays all-1s
  float* xs = lds + wv * (16 * 48);
  const int m0 = tile * 16;
  // stage concat(cf, slf, z) tile [16 x 48] in LDS (irregular gather -> scalar staging)
  for (int idx = lane; idx < 16 * 48; idx += 32) {
    const int r = idx / 48, c = idx % 48, m = m0 + r;
    float v = (c < 15) ? cf[m * 15 + c] : (c == 15 ? slf[m] : z[m * 32 + (c - 16)]);
    xs[idx] = v;
  }
  asm volatile("s_wait_dscnt 0" ::: "memory");
  const int lo = lane & 15, hi = lane >> 4;
  for (int n0 = 0; n0 < Hh; n0 += 16) {
    v8f acc = {};
    acc = gemm_tile(xs, 48, W_enc, 48, n0, acc, lo, hi, 1.0f);
    const float bb = b_enc[n0 + lo];
#pragma unroll
    for (int v = 0; v < 8; ++v) {
      const int row = v + hi * 8;              // D layout: row = v + (lane>>4)*8
      x[(m0 + row) * Hh + n0 + lo] = acc[v] + bb;
    }
  }
}

// ---------------- per-edge scatter: S[dst] += x[src]  (one wave per edge) ----------------
__global__ void edge_scatter_kernel(const float* __restrict__ x, const int* __restrict__ e,
                                    float* __restrict__ S) {
  const int t = blockIdx.x * blockDim.x + threadIdx.x;
  const int lane = t & 31, eid = t >> 5;
  if (eid >= Ee) return;
  const int src = e[eid], dst = e[Ee + eid];
  const float4 v = *(const float4*)(x + (long)src * Hh + lane * 4);
  float* p = S + (long)dst * Hh + lane * 4;
  atomicAdd(p + 0, v.x); atomicAdd(p + 1, v.y);
  atomicAdd(p + 2, v.z); atomicAdd(p + 3, v.w);
}

// ---------------- cluster pool scatter: pooled[cc[n]] += x[n] (one wave per node) -------
__global__ void pool_scatter_kernel(const float* __restrict__ x, const int* __restrict__ cc,
                                    float* __restrict__ pooled) {
  const int t = blockIdx.x * blockDim.x + threadIdx.x;
  const int lane = t & 31, n = t >> 5;
  if (n >= Nn) return;
  const int c = cc[n];
  const float4 v = *(const float4*)(x + (long)n * Hh + lane * 4);
  float* p = pooled + (long)c * Hh + lane * 4;
  atomicAdd(p + 0, v.x); atomicAdd(p + 1, v.y);
  atomicAdd(p + 2, v.z); atomicAdd(p + 3, v.w);
}

// ---------------- fused node update (one wave per 16-node tile) ----------------
// aggr_i = mask_i * ( x_i@Wm_top + b_m + (S_i/deg_i)@Wm_bot )          (exact segment-mean)
// upd    = leaky( x@Wu_x + aggr@Wu_a + (pool*tr/cnt)@Wu_p + b_u ); x += upd  (in place; safe:
// after the algebraic collapse, node i's update reads only x_i, S_i, pooled)
__global__ void layer_update_kernel(float* __restrict__ x, const float* __restrict__ S,
                                    const float* __restrict__ pooled,
                                    const float* __restrict__ deg, const float* __restrict__ ccnt,
                                    const float* __restrict__ tr, const int* __restrict__ cc,
                                    const float* __restrict__ W_m, const float* __restrict__ b_m,
                                    const float* __restrict__ W_u, const float* __restrict__ b_u) {
  extern __shared__ float lds[];
  const int lane = threadIdx.x & 31, wv = threadIdx.x >> 5;
  const int tile = blockIdx.x * (blockDim.x >> 5) + wv;
  if (tile >= NTt) return;                     // wave-uniform exit, no block barriers used
  float* xs  = lds + wv * (4 * 16 * Hh);       // x tile            [16x128]
  float* ss  = xs + 16 * Hh;                   // raw S tile        [16x128]
  float* as_ = ss + 16 * Hh;                   // aggr tile         [16x128]
  float* ps  = as_ + 16 * Hh;                  // raw pooled tile   [16x128]
  const int m0 = tile * 16;

  __builtin_prefetch(W_m, 0, 3);
  __builtin_prefetch(W_u, 0, 3);

  // ---- async-stage raw tiles: memory -> LDS with no VGPR transit (ASYNCcnt) ----
#pragma unroll 4
  for (int r = 0; r < 16; ++r) {
    const int m = m0 + r;
    async_g2l_b128(xs + r * Hh + lane * 4, x + (long)m * Hh + lane * 4);
    async_g2l_b128(ss + r * Hh + lane * 4, S + (long)m * Hh + lane * 4);
    const int c = cc[m];
    async_g2l_b128(ps + r * Hh + lane * 4, pooled + (long)c * Hh + lane * 4);
  }

  // per-lane row scales (A-fragment row M == lane&15 is fixed per lane)
  const int   lo = lane & 15, hi = lane >> 4;
  const float inv_lo    = 1.0f / fmaxf(deg[m0 + lo], 1.0f);     // S/deg scaling
  const int   c_lo      = cc[m0 + lo];
  const float pscale_lo = tr[c_lo] / fmaxf(ccnt[c_lo], 1.0f);   // pooled mean * target ratio
  float rowmask[8];                                             // deg>0 mask per D row
#pragma unroll
  for (int v = 0; v < 8; ++v)
    rowmask[v] = (deg[m0 + v + hi * 8] > 0.0f) ? 1.0f : 0.0f;

  asm volatile("s_wait_asynccnt 0" ::: "memory");

  const float* Wm_top = W_m;                   // multiplies x[dst] part
  const float* Wm_bot = W_m + Hh * Hh;         // multiplies x[src] part

  // ---- aggr ----
  for (int n0 = 0; n0 < Hh; n0 += 16) {
    v8f acc = {};
    acc = gemm_tile(xs, Hh, Wm_top, Hh, n0, acc, lo, hi, 1.0f);
    acc = gemm_tile(ss, Hh, Wm_bot, Hh, n0, acc, lo, hi, inv_lo);
    const float bb = b_m[n0 + lo];
#pragma unroll
    for (int v = 0; v < 8; ++v) {
      const int row = v + hi * 8;
      as_[row * Hh + n0 + lo] = (acc[v] + bb) * rowmask[v];
    }
  }
  asm volatile("s_wait_dscnt 0" ::: "memory");

  // ---- update + residual ----
  const float* Wu_x = W_u;
  const float* Wu_a = W_u + Hh * Hh;
  const float* Wu_p = W_u + 2 * Hh * Hh;
  for (int n0 = 0; n0 < Hh; n0 += 16) {
    v8f acc = {};
    acc = gemm_tile(xs,  Hh, Wu_x, Hh, n0, acc, lo, hi, 1.0f);
    acc = gemm_tile(as_, Hh, Wu_a, Hh, n0, acc, lo, hi, 1.0f);
    acc = gemm_tile(ps,  Hh, Wu_p, Hh, n0, acc, lo, hi, pscale_lo);
    const float bb = b_u[n0 + lo];
#pragma unroll
    for (int v = 0; v < 8; ++v) {
      const int row = v + hi * 8;
      float u = acc[v] + bb;
      u = (u > 0.0f) ? u : NEG_SLOPE * u;
      x[(m0 + row) * Hh + n0 + lo] = xs[row * Hh + n0 + lo] + u;
    }
  }
}

// ---------------- host-side orchestration ----------------
extern "C" void kernel_launch(void* const* d_in, const int* in_sizes, int n_in,
                              void* d_out, int out_size, void* d_ws, size_t ws_size,
                              hipStream_t stream) {
  const float* cf    = (const float*)d_in[0];
  const float* slf   = (const float*)d_in[1];
  const float* z     = (const float*)d_in[2];
  const float* tr    = (const float*)d_in[3];
  const float* W_enc = (const float*)d_in[4];
  const float* b_enc = (const float*)d_in[5];
  const float* W_m   = (const float*)d_in[6];
  const float* b_m   = (const float*)d_in[7];
  const float* W_u   = (const float*)d_in[8];
  const float* b_u   = (const float*)d_in[9];
  const int*   e     = (const int*)d_in[10];
  const int*   cc    = (const int*)d_in[11];
  // d_in[12] = cfb : unused by the reference body

  float* x = (float*)d_out;                    // [N, H], updated in place per layer

  // workspace layout
  float* S      = (float*)d_ws;                // [N, H]   scatter sums
  float* pooled = S + (long)Nn * Hh;           // [BC, H]  cluster sums
  float* degf   = pooled + (long)BCc * Hh;     // [N]      in-degree
  float* ccnt   = degf + Nn;                   // [BC]     cluster sizes

  const long zn_all   = (long)Nn * Hh + (long)BCc * Hh + Nn + BCc;
  const long zn_layer = (long)Nn * Hh + (long)BCc * Hh;

  zero_kernel<<<2048, 256, 0, stream>>>(S, zn_all);
  counts_kernel<<<(Ee + Nn + 255) / 256, 256, 0, stream>>>(e, cc, degf, ccnt);

  // encoder: 8 waves/block, one 16-row tile per wave; LDS = 8 * 16*48 * 4B
  encoder_kernel<<<(NTt + 7) / 8, 256, 8 * 16 * 48 * 4, stream>>>(
      cf, slf, z, W_enc, b_enc, x);

  for (int layer = 0; layer < 3; ++layer) {
    if (layer > 0) zero_kernel<<<2048, 256, 0, stream>>>(S, zn_layer);
    edge_scatter_kernel<<<(Ee * 32) / 256, 256, 0, stream>>>(x, e, S);
    pool_scatter_kernel<<<(Nn * 32 + 255) / 256, 256, 0, stream>>>(x, cc, pooled);
    // 4 waves/block, one tile per wave; LDS = 4 waves * 4 tiles * 16*128 * 4B = 128 KB
    layer_update_kernel<<<(NTt + 3) / 4, 128, 4 * 4 * 16 * Hh * 4, stream>>>(
        x, S, pooled, degf, ccnt, tr, cc, W_m, b_m, W_u, b_u);
  }
}